// Deepseekv3MoE_15161234555174
// MI455X (gfx1250) — compile-verified
//
#include <hip/hip_runtime.h>
#include <hip/hip_bf16.h>
#include <math.h>

// DeepSeek-V3 MoE for MI455X (gfx1250, wave32).
// Memory-bound (~0.9 GB fp32 weights @ 23.3 TB/s => ~39us floor): exact fp32
// V_WMMA_F32_16X16X4_F32 GEMMs, gather-based expert dispatch, and a 2-stage
// async global->LDS pipeline (GLOBAL_LOAD_ASYNC_TO_LDS_B128 / s_wait_asynccnt)
// so HBM streaming overlaps WMMA. Deterministic: per-expert kernels are
// stream-ordered; each output element is owned by one thread per launch.

typedef __attribute__((ext_vector_type(2))) float v2f;
typedef __attribute__((ext_vector_type(8))) float v8f;

constexpr int TT   = 4096;  // tokens
constexpr int HH   = 2048;  // hidden
constexpr int EE   = 32;    // experts
constexpr int II   = 1024;  // expert intermediate
constexpr int ISH  = 2048;  // shared intermediate
constexpr int TOPK = 8;
constexpr int NGRP = 8;
constexpr int TOPKG = 4;
constexpr int EPG  = EE / NGRP;  // experts per group = 4
constexpr float RSCALE = 2.5f;

// GEMM tiling: 64x32 C-tile, K-chunk 32, 8 waves (256 thr), wave owns 16x16.
// LDK = 36: row stride pad => conflict-free v2f fragment reads, 16B-aligned
// rows for b128 async copies (36*4 = 144 = 9*16).
constexpr int BM = 64, BN = 32, BK = 32, LDK = 36;

// ---------------------------------------------------------------------------
// CDNA5 async global->LDS copy path (ASYNCcnt-tracked), with sync fallback.
// Builtin signature (from hipcc diagnostic): param0 is int4* in AS(1).
// ---------------------------------------------------------------------------
#if defined(__HIP_DEVICE_COMPILE__)
#if __has_builtin(__builtin_amdgcn_global_load_async_to_lds_b128) && \
    __has_builtin(__builtin_amdgcn_s_wait_asynccnt)
#define MOE_ASYNC 1
#endif
#endif

typedef __attribute__((__vector_size__(4 * sizeof(int)))) int v4i;
typedef __attribute__((address_space(1))) v4i* gv4i_p;  // global int4*
typedef __attribute__((address_space(3))) v4i* lv4i_p;  // LDS int4*

__device__ __forceinline__ void copy16(const float* g, float* l) {
#ifdef MOE_ASYNC
  __builtin_amdgcn_global_load_async_to_lds_b128((gv4i_p)g, (lv4i_p)l, 0, 0);
#else
  *(float4*)l = *(const float4*)g;
#endif
}

#ifdef MOE_ASYNC
#define WAIT_ASYNC(n) __builtin_amdgcn_s_wait_asynccnt(n)
#else
#define WAIT_ASYNC(n)
#endif

// ---------------------------------------------------------------------------
// Router: one wave32 per token; lane e computes the logit for expert e
// (E == 32 == wave width). Lane 0 runs sigmoid + bias-corrected group top-k
// and appends the token to per-expert gather lists.
// ---------------------------------------------------------------------------
__global__ __launch_bounds__(32)
void router_kernel(const float* __restrict__ x, const float* __restrict__ gw,
                   const float* __restrict__ ebias, int* __restrict__ counts,
                   int* __restrict__ tok_list, float* __restrict__ wt_list)
{
  __shared__ __align__(16) float xs[HH];
  __shared__ float sc[EE];
  const int t = blockIdx.x;
  const int lane = threadIdx.x;

  for (int k = lane * 4; k < HH; k += 32 * 4)
    *(float4*)&xs[k] = *(const float4*)&x[(size_t)t * HH + k];
  __syncthreads();

  const float* wr = gw + (size_t)lane * HH;   // gate_w row for this expert
  float acc = 0.f;
  for (int k = 0; k < HH; k += 4) {
    float4 xv = *(const float4*)&xs[k];
    float4 wv = *(const float4*)&wr[k];
    acc += xv.x * wv.x + xv.y * wv.y + xv.z * wv.z + xv.w * wv.w;
  }
  sc[lane] = 1.f / (1.f + __expf(-acc));      // sigmoid score
  __syncthreads();

  if (lane == 0) {
    float s[EE], swb[EE];
    for (int e = 0; e < EE; ++e) { s[e] = sc[e]; swb[e] = sc[e] + ebias[e]; }
    // group score = max pairwise sum (== top-2 sum over 4 values, tie-safe)
    float gs[NGRP];
    for (int g = 0; g < NGRP; ++g) {
      const float* v = &swb[g * EPG];
      float best = -INFINITY;
      for (int i = 0; i < EPG; ++i)
        for (int j = i + 1; j < EPG; ++j)
          best = fmaxf(best, v[i] + v[j]);
      gs[g] = best;
    }
    bool gsel[NGRP] = {};
    for (int it = 0; it < TOPKG; ++it) {       // stable top-4 groups
      int bi = 0; float bv = -INFINITY;
      for (int g = 0; g < NGRP; ++g)
        if (!gsel[g] && gs[g] > bv) { bv = gs[g]; bi = g; }
      gsel[bi] = true;
    }
    bool esel[EE] = {};
    int idx[TOPK]; float w[TOPK]; float wsum = 0.f;
    for (int it = 0; it < TOPK; ++it) {        // stable top-8 experts
      int bi = 0; float bv = -INFINITY;
      for (int e = 0; e < EE; ++e)
        if (gsel[e / EPG] && !esel[e] && swb[e] > bv) { bv = swb[e]; bi = e; }
      esel[bi] = true; idx[it] = bi; w[it] = s[bi]; wsum += s[bi];
    }
    const float inv = RSCALE / (wsum + 1e-20f);
    for (int it = 0; it < TOPK; ++it) {
      const int e = idx[it];
      const int pos = atomicAdd(&counts[e], 1);
      tok_list[(size_t)e * TT + pos] = t;
      wt_list[(size_t)e * TT + pos] = w[it] * inv;
    }
  }
}

// ---------------------------------------------------------------------------
// Up projection: h[m,n] = silu(X.w1^T) * (X.w3^T), X rows gathered via list.
// Double-buffered async LDS pipeline; dual accumulators share the A fragment.
// Out-of-range rows clamp to row 0 (not skipped) so every wave issues the
// same async-op count per stage; their outputs are never stored.
// ---------------------------------------------------------------------------
__global__ __launch_bounds__(256)
void moe_up_kernel(const float* __restrict__ x, const float* __restrict__ w1,
                   const float* __restrict__ w3, float* __restrict__ hbuf,
                   const int* __restrict__ tok_list, const int* __restrict__ countp,
                   int dense_rows, int Idim)
{
  const int nrows = countp ? countp[0] : dense_rows;
  const int bx = blockIdx.x, by = blockIdx.y;
  if (bx * BM >= nrows) return;

  __shared__ __align__(16) float Xs[2][BM * LDK];
  __shared__ __align__(16) float W1s[2][BN * LDK];
  __shared__ __align__(16) float W3s[2][BN * LDK];

  const int tid = threadIdx.x;
  // X tile: 64x32 floats = 512 b128 chunks -> 2 per thread.
  const int c0 = tid * 2, c1 = tid * 2 + 1;
  const int xr0 = c0 >> 3, xc0 = (c0 & 7) * 4;
  const int xr1 = c1 >> 3, xc1 = (c1 & 7) * 4;
  const int mg0 = bx * BM + xr0, mg1 = bx * BM + xr1;
  const int tk0 = (mg0 < nrows) ? (tok_list ? tok_list[mg0] : mg0) : 0;
  const int tk1 = (mg1 < nrows) ? (tok_list ? tok_list[mg1] : mg1) : 0;
  const float* xrow0 = x + (size_t)tk0 * HH;
  const float* xrow1 = x + (size_t)tk1 * HH;
  // W tiles: 32x32 floats = 256 b128 chunks -> 1 per thread each.
  const int wrw = tid >> 3, wc = (tid & 7) * 4;
  const int ng = by * BN + wrw;                 // always < Idim (exact grid)
  const float* w1row = w1 + (size_t)ng * HH;
  const float* w3row = w3 + (size_t)ng * HH;

  const int lane = tid & 31, wave = tid >> 5;
  const int wm = (wave & 3) * 16, wn = (wave >> 2) * 16;
  const int l15 = lane & 15;
  const int khi = (lane >> 4) * 2;              // A/B lane layout: hi lanes K+2
  const int hi8 = (lane >> 4) * 8;              // C/D lane layout: hi lanes M+8

  v8f acc1 = {}; v8f acc3 = {};

  // stage issue: 4 async b128 per thread per stage
  auto issue = [&](int k0, int b) {
    copy16(xrow0 + k0 + xc0, &Xs[b][xr0 * LDK + xc0]);
    copy16(xrow1 + k0 + xc1, &Xs[b][xr1 * LDK + xc1]);
    copy16(w1row + k0 + wc, &W1s[b][wrw * LDK + wc]);
    copy16(w3row + k0 + wc, &W3s[b][wrw * LDK + wc]);
  };

  issue(0, 0);
  int buf = 0;
  for (int k0 = 0; k0 < HH; k0 += BK) {
    if (k0 + BK < HH) { issue(k0 + BK, buf ^ 1); WAIT_ASYNC(4); }
    else              { WAIT_ASYNC(0); }
    __syncthreads();                       // stage data visible to all waves
#pragma unroll
    for (int kk = 0; kk < BK; kk += 4) {
      const int ka = kk + khi;
      v2f a  = *(const v2f*)&Xs[buf][(wm + l15) * LDK + ka];
      v2f b1 = *(const v2f*)&W1s[buf][(wn + l15) * LDK + ka];
      v2f b3 = *(const v2f*)&W3s[buf][(wn + l15) * LDK + ka];
      acc1 = __builtin_amdgcn_wmma_f32_16x16x4_f32(false, a, false, b1,
                                                   (short)0, acc1, false, false);
      acc3 = __builtin_amdgcn_wmma_f32_16x16x4_f32(false, a, false, b3,
                                                   (short)0, acc3, false, false);
    }
    __syncthreads();                       // all reads of buf done before refill
    buf ^= 1;
  }

#pragma unroll
  for (int r = 0; r < 8; ++r) {
    const int m = bx * BM + wm + hi8 + r;
    if (m < nrows) {
      const float p1 = acc1[r], p3 = acc3[r];
      const float sig = 1.f / (1.f + __expf(-p1));
      hbuf[(size_t)m * Idim + by * BN + wn + l15] = (p1 * sig) * p3;
    }
  }
}

// ---------------------------------------------------------------------------
// Down projection: y = h.w2^T, scaled by combine weight, scattered to out[t].
// accum=0: out = y (shared expert); accum=1: out += wt * y (routed experts).
// ---------------------------------------------------------------------------
__global__ __launch_bounds__(256)
void moe_down_kernel(const float* __restrict__ hbuf, const float* __restrict__ w2,
                     float* __restrict__ out, const int* __restrict__ tok_list,
                     const float* __restrict__ wt_list, const int* __restrict__ countp,
                     int dense_rows, int Idim, int accum)
{
  const int nrows = countp ? countp[0] : dense_rows;
  const int bx = blockIdx.x, by = blockIdx.y;
  if (bx * BM >= nrows) return;

  __shared__ __align__(16) float As[2][BM * LDK];
  __shared__ __align__(16) float Bs[2][BN * LDK];

  const int tid = threadIdx.x;
  const int c0 = tid * 2, c1 = tid * 2 + 1;
  const int ar0 = c0 >> 3, ac0 = (c0 & 7) * 4;
  const int ar1 = c1 >> 3, ac1 = (c1 & 7) * 4;
  const int mg0 = bx * BM + ar0, mg1 = bx * BM + ar1;
  const float* arow0 = hbuf + (size_t)((mg0 < nrows) ? mg0 : 0) * Idim;
  const float* arow1 = hbuf + (size_t)((mg1 < nrows) ? mg1 : 0) * Idim;
  const int wrw = tid >> 3, wc = (tid & 7) * 4;
  const int ng = by * BN + wrw;                 // always < HH (exact grid)
  const float* brow = w2 + (size_t)ng * Idim;

  const int lane = tid & 31, wave = tid >> 5;
  const int wm = (wave & 3) * 16, wn = (wave >> 2) * 16;
  const int l15 = lane & 15;
  const int khi = (lane >> 4) * 2;
  const int hi8 = (lane >> 4) * 8;

  v8f acc = {};

  auto issue = [&](int k0, int b) {             // 3 async b128 per thread
    copy16(arow0 + k0 + ac0, &As[b][ar0 * LDK + ac0]);
    copy16(arow1 + k0 + ac1, &As[b][ar1 * LDK + ac1]);
    copy16(brow + k0 + wc, &Bs[b][wrw * LDK + wc]);
  };

  issue(0, 0);
  int buf = 0;
  for (int k0 = 0; k0 < Idim; k0 += BK) {
    if (k0 + BK < Idim) { issue(k0 + BK, buf ^ 1); WAIT_ASYNC(3); }
    else                { WAIT_ASYNC(0); }
    __syncthreads();
#pragma unroll
    for (int kk = 0; kk < BK; kk += 4) {
      const int ka = kk + khi;
      v2f a = *(const v2f*)&As[buf][(wm + l15) * LDK + ka];
      v2f b = *(const v2f*)&Bs[buf][(wn + l15) * LDK + ka];
      acc = __builtin_amdgcn_wmma_f32_16x16x4_f32(false, a, false, b,
                                                  (short)0, acc, false, false);
    }
    __syncthreads();
    buf ^= 1;
  }

#pragma unroll
  for (int r = 0; r < 8; ++r) {
    const int m = bx * BM + wm + hi8 + r;
    if (m < nrows) {
      const int t = tok_list ? tok_list[m] : m;
      const float sc = wt_list ? wt_list[m] : 1.f;
      const size_t o = (size_t)t * HH + by * BN + wn + l15;
      const float y = acc[r] * sc;
      if (accum) out[o] += y; else out[o] = y;
    }
  }
}

// ---------------------------------------------------------------------------
extern "C" void kernel_launch(void* const* d_in, const int* in_sizes, int n_in,
                              void* d_out, int out_size, void* d_ws, size_t ws_size,
                              hipStream_t stream)
{
  const float* x     = (const float*)d_in[0];  // [T,H]
  const float* gw    = (const float*)d_in[1];  // [E,H]
  const float* ebias = (const float*)d_in[2];  // [E]
  const float* w1    = (const float*)d_in[3];  // [E,I,H]
  const float* w3    = (const float*)d_in[4];  // [E,I,H]
  const float* w2    = (const float*)d_in[5];  // [E,H,I]
  const float* sw1   = (const float*)d_in[6];  // [IS,H]
  const float* sw3   = (const float*)d_in[7];  // [IS,H]
  const float* sw2   = (const float*)d_in[8];  // [H,IS]
  float* out = (float*)d_out;                  // [T,H] fp32

  // workspace layout (~35 MB): counts | tok_list | wt_list | hbuf(T*IS f32)
  char* ws = (char*)d_ws;
  int*   counts   = (int*)ws;
  int*   tok_list = (int*)(ws + 256);
  float* wt_list  = (float*)(ws + 256 + (size_t)EE * TT * 4);
  float* hbuf     = (float*)(ws + 256 + 2 * (size_t)EE * TT * 4);

  (void)hipMemsetAsync(counts, 0, EE * sizeof(int), stream);
  router_kernel<<<TT, 32, 0, stream>>>(x, gw, ebias, counts, tok_list, wt_list);

  // shared expert (dense over all tokens) writes out
  moe_up_kernel<<<dim3(TT / BM, ISH / BN), 256, 0, stream>>>(
      x, sw1, sw3, hbuf, nullptr, nullptr, TT, ISH);
  moe_down_kernel<<<dim3(TT / BM, HH / BN), 256, 0, stream>>>(
      hbuf, sw2, out, nullptr, nullptr, nullptr, TT, ISH, /*accum=*/0);

  // routed experts: gathered token lists, stream-ordered accumulation into out
  for (int e = 0; e < EE; ++e) {
    const float* w1e = w1 + (size_t)e * II * HH;
    const float* w3e = w3 + (size_t)e * II * HH;
    const float* w2e = w2 + (size_t)e * HH * II;
    moe_up_kernel<<<dim3(TT / BM, II / BN), 256, 0, stream>>>(
        x, w1e, w3e, hbuf, tok_list + (size_t)e * TT, counts + e, TT, II);
    moe_down_kernel<<<dim3(TT / BM, HH / BN), 256, 0, stream>>>(
        hbuf, w2e, out, tok_list + (size_t)e * TT, wt_list + (size_t)e * TT,
        counts + e, TT, II, /*accum=*/1);
  }

  (void)in_sizes; (void)n_in; (void)out_size; (void)ws_size;
}